// TtMistralAttention_3221225472101
// MI455X (gfx1250) — compile-verified
//
#include <hip/hip_runtime.h>
#include <hip/hip_bf16.h>

// ---------------------------------------------------------------------------
// Mistral decode attention for MI455X (gfx1250), fp32, bandwidth-bound.
// B=32, H=32, KVH=8, D=128, HID=4096, WIN=4096, PAD_L=3008
//
// Roofline: ~0.95 GB read (wqkv 100MB + wo 67MB + KV slice 788MB) vs 4.3
// GFLOP -> pure HBM-bound (~41us floor @ 23.3 TB/s). Strategy:
//  * WMMA f32 GEMMs, split-K x16 (-> 4-6k waves) and 4-step load batching so
//    each SIMD keeps ~9KB of weight stream in flight (latency coverage).
//  * KV stream is non-temporal so wqkv+wo (167MB) stay L2-resident (192MB)
//    across graph replays.
// ---------------------------------------------------------------------------

typedef float v2f __attribute__((ext_vector_type(2)));
typedef float v4f __attribute__((ext_vector_type(4)));
typedef float v8f __attribute__((ext_vector_type(8)));

#define B_     32
#define H_     32
#define KVH_   8
#define D_     128
#define HID_   4096
#define WIN_   4096
#define QKVN_  6144   // (H + 2*KVH) * D
#define PADL_  3008   // nearest_32(3001)
#define KSLICES_ 16

// ---------------------------------------------------------------------------
// Split-K GEMM: P[slice][32][N] = A[32, kb:kb+KLEN] @ W[kb:kb+KLEN, N] with
// V_WMMA_F32_16X16X4_F32. One wave per 16-wide N tile; both 16-row M tiles
// per wave so the B (weight) fragment is loaded exactly once. K/N are
// template constants so all strides fold into immediate load offsets, and
// 4 k-steps of fragments are fetched before any WMMA so the loads form one
// clause with staged (not full-drain) waits.
//
// WMMA f32 16x16x4 VGPR layouts (ISA 7.12.2):
//   A (16x4): lanes 0-15 hold M=lane, v.x=K0 v.y=K1; lanes 16-31: K2/K3
//   B (4x16): lanes 0-15 hold N=lane, v.x=K0 v.y=K1; lanes 16-31: K2/K3
//   C/D:      VGPR r: lanes 0-15 -> M=r, lanes 16-31 -> M=r+8; N=lane%16
// ---------------------------------------------------------------------------
template <int K, int N, int KLEN>
__global__ void gemm32_f32_wmma_splitk(const float* __restrict__ A,
                                       const float* __restrict__ W,
                                       float* __restrict__ P) {
  const int lane  = threadIdx.x & 31;
  const int wave  = threadIdx.x >> 5;
  const int n0    = (blockIdx.x * (blockDim.x >> 5) + wave) * 16;
  const int slice = blockIdx.y;
  if (n0 >= N) return;                 // wave-uniform: EXEC stays all-ones

  const int half = lane >> 4;          // 0: K pair {0,1}, 1: K pair {2,3}
  const int l16  = lane & 15;
  const int koff = half * 2;
  const int kbeg = slice * KLEN;

  const float* Ap0 = A + (size_t)l16 * K + kbeg + koff;   // M rows 0..15
  const float* Ap1 = Ap0 + (size_t)16 * K;                // M rows 16..31
  const float* Wp  = W + (size_t)(kbeg + koff) * N + n0 + l16;

  v8f c0 = {};
  v8f c1 = {};

  for (int k = 0; k < KLEN; k += 16) {
    v2f a0[4], a1[4], bb[4];
#pragma unroll
    for (int u = 0; u < 4; ++u) {
      a0[u].x = Ap0[4 * u];
      a0[u].y = Ap0[4 * u + 1];
      a1[u].x = Ap1[4 * u];
      a1[u].y = Ap1[4 * u + 1];
      bb[u].x = Wp[(size_t)(4 * u) * N];
      bb[u].y = Wp[(size_t)(4 * u + 1) * N];
    }
#pragma unroll
    for (int u = 0; u < 4; ++u) {
      c0 = __builtin_amdgcn_wmma_f32_16x16x4_f32(false, a0[u], false, bb[u],
                                                 (short)0, c0, false, false);
      c1 = __builtin_amdgcn_wmma_f32_16x16x4_f32(false, a1[u], false, bb[u],
                                                 (short)0, c1, false, false);
    }
    Ap0 += 16; Ap1 += 16; Wp += (size_t)16 * N;
  }

  float* out = P + (size_t)slice * B_ * N;
#pragma unroll
  for (int r = 0; r < 8; ++r) {
    const int row = r + half * 8;
    out[(size_t)row * N + n0 + l16]        = c0[r];
    out[(size_t)(row + 16) * N + n0 + l16] = c1[r];
  }
}

// ---------------------------------------------------------------------------
// Deterministic split-K reduction: dst[i] = sum_s P[s][i]
// ---------------------------------------------------------------------------
__global__ void reduce_slices(const float* __restrict__ P,
                              float* __restrict__ dst, int total) {
  const int i = blockIdx.x * blockDim.x + threadIdx.x;
  if (i >= total) return;
  float s = 0.f;
#pragma unroll
  for (int w = 0; w < KSLICES_; ++w) s += P[(size_t)w * total + i];
  dst[i] = s;
}

// ---------------------------------------------------------------------------
// RoPE on the q (cols 0..H*D) and k (cols H*D..(H+KVH)*D) sections of xqkv,
// in place. Each thread owns a (d, d+64) pair -> no read/write race.
// ---------------------------------------------------------------------------
__global__ void rope_kernel(float* __restrict__ xqkv,
                            const float* __restrict__ cosc,
                            const float* __restrict__ sinc,
                            const int* __restrict__ start_pos) {
  const int idx = blockIdx.x * blockDim.x + threadIdx.x;  // 32 * 40 * 64
  if (idx >= B_ * (H_ + KVH_) * (D_ / 2)) return;
  const int d    = idx & 63;
  const int hb   = idx >> 6;
  const int head = hb % (H_ + KVH_);   // 0..31 q heads, 32..39 k heads
  const int b    = hb / (H_ + KVH_);
  const int base = b * QKVN_ + head * D_;

  const int s = start_pos[0];
  const float lo = xqkv[base + d];
  const float hi = xqkv[base + d + 64];
  const float cl = cosc[s * D_ + d];
  const float ch = cosc[s * D_ + d + 64];
  const float sl = sinc[s * D_ + d];
  const float sh = sinc[s * D_ + d + 64];
  // out = u*cos + rotate_half(u)*sin ; rotate_half = [-u_hi, u_lo]
  xqkv[base + d]      = lo * cl - hi * sl;
  xqkv[base + d + 64] = hi * ch + lo * sh;
}

// ---------------------------------------------------------------------------
// GQA flash-decode attention. One block per (b, kvh); 8 waves split the
// L=3008 window. Coalesced non-temporal 512B row loads (the 788MB KV stream
// must not evict L2-resident weights), in-wave shuffle reductions (DS pipe,
// co-executes with VALU), online softmax, cross-wave merge in LDS. The k/v
// row at current_pos comes from the RoPE'd xqkv (inputs never mutated); the
// L range is split around current_pos so the hot loop carries no selects.
// ---------------------------------------------------------------------------
__global__ void attn_kernel(const float* __restrict__ xqkv,
                            const float* __restrict__ cache_k,
                            const float* __restrict__ cache_v,
                            const int* __restrict__ start_pos,
                            const int* __restrict__ current_pos,
                            float* __restrict__ ctx) {
  const int b    = blockIdx.x >> 3;
  const int kvh  = blockIdx.x & 7;
  const int lane = threadIdx.x & 31;
  const int wave = threadIdx.x >> 5;
  const int sp   = start_pos[0];
  const int cur  = current_pos[0];

  __shared__ float q_s[4][D_];
  __shared__ float sm[8][4];
  __shared__ float ssum[8][4];
  __shared__ float sacc[8][4][D_];

  const float scale = 0.08838834764831845f;  // 1/sqrt(128)
  for (int i = threadIdx.x; i < 4 * D_; i += blockDim.x) {
    const int g = i >> 7, d = i & 127;
    q_s[g][d] = xqkv[b * QKVN_ + (kvh * 4 + g) * D_ + d] * scale;
  }
  __syncthreads();

  const float* kbase = cache_k + (size_t)(b * KVH_ + kvh) * WIN_ * D_;
  const float* vbase = cache_v + (size_t)(b * KVH_ + kvh) * WIN_ * D_;

  float qreg[4][4];
#pragma unroll
  for (int g = 0; g < 4; ++g)
#pragma unroll
    for (int j = 0; j < 4; ++j) qreg[g][j] = q_s[g][lane * 4 + j];

  float m[4], su[4], acc[4][4];
#pragma unroll
  for (int g = 0; g < 4; ++g) {
    m[g] = -1e30f; su[g] = 0.f;
    acc[g][0] = acc[g][1] = acc[g][2] = acc[g][3] = 0.f;
  }

  // one online-softmax step for all 4 query heads of this KV head
  auto update = [&](const v4f kr, const v4f vr, const float msk) {
#pragma unroll
    for (int g = 0; g < 4; ++g) {
      float t = qreg[g][0] * kr.x + qreg[g][1] * kr.y +
                qreg[g][2] * kr.z + qreg[g][3] * kr.w;
#pragma unroll
      for (int off = 16; off > 0; off >>= 1) t += __shfl_xor(t, off, 32);
      t += msk;
      const float mo   = m[g];
      const float mn   = fmaxf(mo, t);
      const float corr = __expf(mo - mn);
      const float p    = __expf(t - mn);
      m[g]  = mn;
      su[g] = su[g] * corr + p;
      acc[g][0] = acc[g][0] * corr + p * vr.x;
      acc[g][1] = acc[g][1] * corr + p * vr.y;
      acc[g][2] = acc[g][2] * corr + p * vr.z;
      acc[g][3] = acc[g][3] * corr + p * vr.w;
    }
  };

  auto run = [&](const int s, const int e) {
    for (int l = s; l < e; ++l) {
      if (l + 16 < e) {
        __builtin_prefetch(&kbase[(size_t)(l + 16) * D_ + lane * 4], 0, 0);
        __builtin_prefetch(&vbase[(size_t)(l + 16) * D_ + lane * 4], 0, 0);
      }
      const v4f kr = __builtin_nontemporal_load(
          (const v4f*)&kbase[(size_t)l * D_ + lane * 4]);
      const v4f vr = __builtin_nontemporal_load(
          (const v4f*)&vbase[(size_t)l * D_ + lane * 4]);
      update(kr, vr, (l <= sp) ? 0.f : -1e9f);
    }
  };

  const int chunk = PADL_ / 8;          // 376, exact
  const int l0 = wave * chunk, l1 = l0 + chunk;

  // [l0, min(l1, cur)) from cache, cur (if owned) from fresh k/v, rest cache
  const int cut = (cur < l0) ? l0 : ((cur < l1) ? cur : l1);
  run(l0, cut);
  if (cur >= l0 && cur < l1) {
    const v4f kn = *(const v4f*)&xqkv[b * QKVN_ + H_ * D_ + kvh * D_ + lane * 4];
    const v4f vn = *(const v4f*)&xqkv[b * QKVN_ + (H_ + KVH_) * D_ + kvh * D_ + lane * 4];
    update(kn, vn, (cur <= sp) ? 0.f : -1e9f);
    run(cur + 1, l1);
  } else {
    run(cut, l1);
  }

#pragma unroll
  for (int g = 0; g < 4; ++g) {
    if (lane == 0) { sm[wave][g] = m[g]; ssum[wave][g] = su[g]; }
#pragma unroll
    for (int j = 0; j < 4; ++j) sacc[wave][g][lane * 4 + j] = acc[g][j];
  }
  __syncthreads();

  for (int idx = threadIdx.x; idx < 4 * D_; idx += blockDim.x) {
    const int g = idx >> 7, d = idx & 127;
    float M = -1e30f;
#pragma unroll
    for (int w = 0; w < 8; ++w) M = fmaxf(M, sm[w][g]);
    float S = 0.f, Av = 0.f;
#pragma unroll
    for (int w = 0; w < 8; ++w) {
      const float e = __expf(sm[w][g] - M);
      S  += ssum[w][g] * e;
      Av += sacc[w][g][d] * e;
    }
    ctx[(size_t)b * HID_ + (kvh * 4 + g) * D_ + d] = Av / S;
  }
}

// ---------------------------------------------------------------------------
// Launch: K1 split-K qkv gemm + reduce -> rope -> attention -> split-K wo
// gemm + reduce. inputs: x, wqkv, wo, cache_k, cache_v, cos_cached,
// sin_cached, attn_mask, start_pos, current_pos
// ---------------------------------------------------------------------------
extern "C" void kernel_launch(void* const* d_in, const int* in_sizes, int n_in,
                              void* d_out, int out_size, void* d_ws, size_t ws_size,
                              hipStream_t stream) {
  const float* x       = (const float*)d_in[0];
  const float* wqkv    = (const float*)d_in[1];
  const float* wo      = (const float*)d_in[2];
  const float* cache_k = (const float*)d_in[3];
  const float* cache_v = (const float*)d_in[4];
  const float* cosc    = (const float*)d_in[5];
  const float* sinc    = (const float*)d_in[6];
  const int*   sp      = (const int*)d_in[8];
  const int*   cp      = (const int*)d_in[9];
  float*       out     = (float*)d_out;

  float* xqkv = (float*)d_ws;            // 32 x 6144
  float* ctx  = xqkv + B_ * QKVN_;       // 32 x 4096
  float* part = ctx + B_ * HID_;         // KSLICES_ x 32 x 6144 (reused for wo)

  // K1: xqkv = x @ wqkv  (N=6144, K=4096), split-K x16 -> 6144 waves
  gemm32_f32_wmma_splitk<HID_, QKVN_, HID_ / KSLICES_>
      <<<dim3(QKVN_ / 64, KSLICES_), 128, 0, stream>>>(x, wqkv, part);
  reduce_slices<<<(B_ * QKVN_) / 256, 256, 0, stream>>>(part, xqkv, B_ * QKVN_);

  // K2: RoPE on q and k sections
  rope_kernel<<<(B_ * (H_ + KVH_) * (D_ / 2)) / 256, 256, 0, stream>>>(
      xqkv, cosc, sinc, sp);

  // K3: attention, one block per (b, kvh)
  attn_kernel<<<B_ * KVH_, 256, 0, stream>>>(xqkv, cache_k, cache_v, sp, cp, ctx);

  // K4: out = ctx @ wo   (N=4096, K=4096), split-K x16 -> 4096 waves
  gemm32_f32_wmma_splitk<HID_, HID_, HID_ / KSLICES_>
      <<<dim3(HID_ / 64, KSLICES_), 128, 0, stream>>>(ctx, wo, part);
  reduce_slices<<<(B_ * HID_) / 256, 256, 0, stream>>>(part, out, B_ * HID_);
}